// TDrumorGCN_12824772345978
// MI455X (gfx1250) — compile-verified
//
#include <hip/hip_runtime.h>

#define IN_F  128
#define HID_F 64
#define OUT_F 64
#define CAT_F (OUT_F + HID_F)   // 128, final per-node feature width

typedef float v2f __attribute__((ext_vector_type(2)));
typedef float v8f __attribute__((ext_vector_type(8)));

#if __has_builtin(__builtin_amdgcn_wmma_f32_16x16x4_f32)
#define USE_WMMA_F32 1
#else
#define USE_WMMA_F32 0
#endif

#if USE_WMMA_F32
__device__ __forceinline__ v8f wmma_f32(v2f a, v2f b, v8f c) {
  // (neg_a, A, neg_b, B, c_mod, C, reuse_a, reuse_b) — same 8-arg pattern as
  // the codegen-confirmed 16x16x32_f16 builtin.
  return __builtin_amdgcn_wmma_f32_16x16x4_f32(false, a, false, b, (short)0, c, false, false);
}
#endif

// ---------------------------------------------------------------- small ops
__global__ void k_deg(const long long* __restrict__ dst, float* __restrict__ degf, int E) {
  int e = blockIdx.x * blockDim.x + threadIdx.x;
  if (e < E) atomicAdd(&degf[(int)dst[e]], 1.0f);
}

__global__ void k_dinv(const float* __restrict__ degf, float* __restrict__ dinv, int N) {
  int n = blockIdx.x * blockDim.x + threadIdx.x;
  if (n < N) dinv[n] = rsqrtf(degf[n] + 1.0f);
}

__global__ void k_rootmin(const long long* __restrict__ batch, int* __restrict__ rootidx, int N) {
  int n = blockIdx.x * blockDim.x + threadIdx.x;
  if (n < N) atomicMin(&rootidx[(int)batch[n]], n);
}

__global__ void k_rootnode(const long long* __restrict__ batch, const int* __restrict__ rootidx,
                           int* __restrict__ rootnode, int N) {
  int n = blockIdx.x * blockDim.x + threadIdx.x;
  if (n < N) {
    int r = rootidx[(int)batch[n]];
    rootnode[n] = (r > N - 1) ? (N - 1) : r;
  }
}

// ------------------------------------------------- GEMM1: h1_pre = x @ W1
// One wave per 16-row x 64-col output tile (4 WMMA accumulators).
__global__ void k_gemm1(const float* __restrict__ x, const float* __restrict__ W,
                        float* __restrict__ out, int N) {
  const int wave = threadIdx.x >> 5;
  const int lane = threadIdx.x & 31;
  const int tile = blockIdx.x * 8 + wave;
  const int row0 = tile * 16;
  if (row0 >= N) return;                 // wave-uniform exit (EXEC all-1 for WMMA)
  const int lm = lane & 15;
  const int hi = lane >> 4;              // 0: K=+0,+1   1: K=+2,+3
  int r = row0 + lm; if (r > N - 1) r = N - 1;
  const float* __restrict__ arow = x + (size_t)r * IN_F + 2 * hi;

  v8f acc[4] = {};
#if USE_WMMA_F32
  for (int k = 0; k < IN_F; k += 4) {
    v2f a; a.x = arow[k]; a.y = arow[k + 1];
    const float* bp = W + (size_t)(k + 2 * hi) * HID_F + lm;
#pragma unroll
    for (int t = 0; t < 4; ++t) {
      v2f b; b.x = bp[t * 16]; b.y = bp[HID_F + t * 16];
      acc[t] = wmma_f32(a, b, acc[t]);
    }
  }
#else
#pragma unroll
  for (int t = 0; t < 4; ++t)
    for (int i = 0; i < 8; ++i) {
      int row = row0 + hi * 8 + i; if (row > N - 1) row = N - 1;
      float s = 0.f;
      for (int k = 0; k < IN_F; ++k) s += x[(size_t)row * IN_F + k] * W[(size_t)k * HID_F + t * 16 + lm];
      acc[t][i] = s;
    }
#endif
#pragma unroll
  for (int t = 0; t < 4; ++t)
#pragma unroll
    for (int i = 0; i < 8; ++i) {
      int row = row0 + hi * 8 + i;
      if (row < N) out[(size_t)row * HID_F + t * 16 + lm] = acc[t][i];
    }
}

// ------------------- GEMM2: h2_pre = relu([h1 | x[root]]) @ W2   (K = 192)
__global__ void k_gemm2(const float* __restrict__ h1, const float* __restrict__ x,
                        const int* __restrict__ rootnode, const float* __restrict__ W2,
                        float* __restrict__ out, int N) {
  const int wave = threadIdx.x >> 5;
  const int lane = threadIdx.x & 31;
  const int tile = blockIdx.x * 8 + wave;
  const int row0 = tile * 16;
  if (row0 >= N) return;
  const int lm = lane & 15;
  const int hi = lane >> 4;
  int r = row0 + lm; if (r > N - 1) r = N - 1;
  const float* __restrict__ h1row = h1 + (size_t)r * HID_F;
  const float* __restrict__ xrow  = x + (size_t)rootnode[r] * IN_F;

  v8f acc[4] = {};
#if USE_WMMA_F32
  for (int k = 0; k < HID_F + IN_F; k += 4) {
    const int kk = k + 2 * hi;           // region never straddles k=64 (step 4)
    float a0, a1;
    if (k < HID_F) { a0 = h1row[kk];          a1 = h1row[kk + 1]; }
    else           { a0 = xrow[kk - HID_F];   a1 = xrow[kk - HID_F + 1]; }
    v2f a; a.x = fmaxf(a0, 0.f); a.y = fmaxf(a1, 0.f);
    const float* bp = W2 + (size_t)kk * OUT_F + lm;
#pragma unroll
    for (int t = 0; t < 4; ++t) {
      v2f b; b.x = bp[t * 16]; b.y = bp[OUT_F + t * 16];
      acc[t] = wmma_f32(a, b, acc[t]);
    }
  }
#else
#pragma unroll
  for (int t = 0; t < 4; ++t)
    for (int i = 0; i < 8; ++i) {
      int row = row0 + hi * 8 + i; if (row > N - 1) row = N - 1;
      const float* hr = h1 + (size_t)row * HID_F;
      const float* xr = x + (size_t)rootnode[row] * IN_F;
      float s = 0.f;
      for (int k = 0; k < HID_F + IN_F; ++k) {
        float av = (k < HID_F) ? hr[k] : xr[k - HID_F];
        s += fmaxf(av, 0.f) * W2[(size_t)k * OUT_F + t * 16 + lm];
      }
      acc[t][i] = s;
    }
#endif
#pragma unroll
  for (int t = 0; t < 4; ++t)
#pragma unroll
    for (int i = 0; i < 8; ++i) {
      int row = row0 + hi * 8 + i;
      if (row < N) out[(size_t)row * OUT_F + t * 16 + lm] = acc[t][i];
    }
}

// -------------------------------------------- edge scatter (message passing)
// One thread per (edge, feature): 64 consecutive lanes gather one source row
// (256B coalesced) and scatter-add into agg[dst] with f32 global atomics.
__global__ void k_scatter(const float* __restrict__ hpre, const long long* __restrict__ src,
                          const long long* __restrict__ dst, const float* __restrict__ dinv,
                          float* __restrict__ agg, int E) {
  long long t = (long long)blockIdx.x * blockDim.x + threadIdx.x;
  int e = (int)(t >> 6);
  if (e >= E) return;
  int f = (int)(t & 63);
  int s = (int)src[e];
  int d = (int)dst[e];
  float norm = dinv[s] * dinv[d];
  atomicAdd(&agg[(size_t)d * 64 + f], hpre[(size_t)s * 64 + f] * norm);
}

// h1 = agg + h1_pre*dinv^2 + b1  (pre-ReLU value; also serves as root2 source)
__global__ void k_combine1(const float* __restrict__ agg, const float* __restrict__ hpre,
                           const float* __restrict__ dinv, const float* __restrict__ b1,
                           float* __restrict__ h1, int N) {
  int t = blockIdx.x * blockDim.x + threadIdx.x;
  if (t >= N * HID_F) return;
  int n = t >> 6, f = t & 63;
  float di = dinv[n];
  h1[t] = agg[t] + hpre[t] * di * di + b1[f];
}

__global__ void k_cnt(const long long* __restrict__ batch, float* __restrict__ cnt, int N) {
  int n = blockIdx.x * blockDim.x + threadIdx.x;
  if (n < N) atomicAdd(&cnt[(int)batch[n]], 1.0f);
}

// final per-node features [relu(conv2) | h1[root]] scatter-summed per graph
__global__ void k_pool(const float* __restrict__ agg2, const float* __restrict__ h2pre,
                       const float* __restrict__ dinv, const float* __restrict__ b2,
                       const float* __restrict__ h1, const int* __restrict__ rootnode,
                       const long long* __restrict__ batch, float* __restrict__ sums, int N) {
  int t = blockIdx.x * blockDim.x + threadIdx.x;
  if (t >= N * CAT_F) return;
  int n = t >> 7, f = t & 127;
  int g = (int)batch[n];
  float v;
  if (f < OUT_F) {
    float di = dinv[n];
    v = agg2[(size_t)n * OUT_F + f] + h2pre[(size_t)n * OUT_F + f] * di * di + b2[f];
    v = fmaxf(v, 0.f);
  } else {
    v = h1[(size_t)rootnode[n] * HID_F + (f - OUT_F)];
  }
  atomicAdd(&sums[(size_t)g * CAT_F + f], v);
}

__global__ void k_out(const float* __restrict__ sums, const float* __restrict__ cnt,
                      float* __restrict__ out, int G) {
  int t = blockIdx.x * blockDim.x + threadIdx.x;
  if (t >= G * CAT_F) return;
  out[t] = sums[t] / fmaxf(cnt[t >> 7], 1.0f);
}

// ---------------------------------------------------------------------------
extern "C" void kernel_launch(void* const* d_in, const int* in_sizes, int n_in,
                              void* d_out, int out_size, void* d_ws, size_t ws_size,
                              hipStream_t stream) {
  const float*     x     = (const float*)d_in[0];
  const long long* ei    = (const long long*)d_in[1];
  const long long* batch = (const long long*)d_in[2];
  const float*     W1    = (const float*)d_in[3];
  const float*     b1    = (const float*)d_in[4];
  const float*     W2    = (const float*)d_in[5];
  const float*     b2    = (const float*)d_in[6];

  const int N = in_sizes[0] / IN_F;
  const int E = in_sizes[1] / 2;
  const int G = out_size / CAT_F;
  const long long* src = ei;
  const long long* dst = ei + E;

  // workspace layout (floats)
  float* ws   = (float*)d_ws;
  size_t nf   = (size_t)N * 64;
  float* bufA = ws;                 // h_pre (conv1 then conv2)
  float* bufB = ws + nf;            // agg (scatter target, zeroed per conv)
  float* bufC = ws + 2 * nf;        // h1 (pre-ReLU, kept for root2)
  float* degf = ws + 3 * nf;
  float* dinv = degf + N;
  int*   rootnode = (int*)(dinv + N);
  int*   rootidx  = rootnode + N;
  float* sums = (float*)(rootidx + G);
  float* cnt  = sums + (size_t)G * CAT_F;

  hipMemsetAsync(degf, 0, (size_t)N * sizeof(float), stream);
  hipMemsetAsync(bufB, 0, nf * sizeof(float), stream);
  hipMemsetAsync(rootidx, 0x7F, (size_t)G * sizeof(int), stream);   // ~INT_MAX
  hipMemsetAsync(sums, 0, (size_t)G * CAT_F * sizeof(float), stream);
  hipMemsetAsync(cnt, 0, (size_t)G * sizeof(float), stream);

  const int T = 256;
  k_deg     <<<(E + T - 1) / T, T, 0, stream>>>(dst, degf, E);
  k_dinv    <<<(N + T - 1) / T, T, 0, stream>>>(degf, dinv, N);
  k_rootmin <<<(N + T - 1) / T, T, 0, stream>>>(batch, rootidx, N);
  k_rootnode<<<(N + T - 1) / T, T, 0, stream>>>(batch, rootidx, rootnode, N);

  const int tiles   = (N + 15) / 16;
  const int gblocks = (tiles + 7) / 8;        // 8 waves (16-row tiles) per block
  k_gemm1<<<gblocks, 256, 0, stream>>>(x, W1, bufA, N);

  const long long tot = (long long)E * 64;
  const unsigned sblocks = (unsigned)((tot + T - 1) / T);
  k_scatter<<<sblocks, T, 0, stream>>>(bufA, src, dst, dinv, bufB, E);
  k_combine1<<<(N * HID_F + T - 1) / T, T, 0, stream>>>(bufB, bufA, dinv, b1, bufC, N);

  k_gemm2<<<gblocks, 256, 0, stream>>>(bufC, x, rootnode, W2, bufA, N);
  hipMemsetAsync(bufB, 0, nf * sizeof(float), stream);   // after combine1 read
  k_scatter<<<sblocks, T, 0, stream>>>(bufA, src, dst, dinv, bufB, E);

  k_cnt <<<(N + T - 1) / T, T, 0, stream>>>(batch, cnt, N);
  k_pool<<<(N * CAT_F + T - 1) / T, T, 0, stream>>>(bufB, bufA, dinv, b2, bufC, rootnode,
                                                    batch, sums, N);
  k_out <<<(G * CAT_F + T - 1) / T, T, 0, stream>>>(sums, cnt, (float*)d_out, G);
}